// FCM_1_42494406426874
// MI455X (gfx1250) — compile-verified
//
#include <hip/hip_runtime.h>

// Problem constants (from reference): A=8, PILE=4, COL=128, T=256, L=PILE-2=2
#define AN   8
#define COLN 128
#define TN   256
#define LN   2

typedef float v2f __attribute__((ext_vector_type(2)));
typedef float v8f __attribute__((ext_vector_type(8)));

#define WMMA_F32(a, b, c) \
  __builtin_amdgcn_wmma_f32_16x16x4_f32(false, (a), false, (b), (short)0, (c), false, false)

// ---------------------------------------------------------------------------
// Batched GEMM via V_WMMA_F32_16X16X4_F32 (wave32).
//   C[a][m][n] = sum_k A[a][m][k] * W[a][n][k] + bias[n]
// One wave computes a 16x64 output strip: 4 f32 accumulators share each
// A-operand load (5 loads : 4 WMMAs), and the K-loop is software-pipelined
// (loads for step k+4 issued before the WMMAs of step k) so the compiler can
// overlap L2 latency with matrix math instead of a full s_wait per WMMA.
// Grid is sized so every wave owns a valid strip -> EXEC all ones (WMMA req).
// ---------------------------------------------------------------------------
__global__ __launch_bounds__(128) void gemm_wmma_f32(
    const float* __restrict__ Abase, long long strideA,
    const float* __restrict__ Wbase, long long strideW,
    const float* __restrict__ bias,          // nullptr => no bias
    float* __restrict__ Cbase, long long strideC,
    int N, int K, int tilesM, int tilesN64)
{
  const int wave = blockIdx.x * 4 + (threadIdx.x >> 5);
  const int lane = threadIdx.x & 31;
  const int per  = tilesM * tilesN64;
  const int a    = wave / per;
  const int r    = wave - a * per;
  const int tm   = r / tilesN64;
  const int tn   = r - tm * tilesN64;      // strip index, 64 columns each
  const int lr   = lane & 15;
  const int half = lane >> 4;

  // ISA 7.12.2 (32-bit A 16x4): lane holds row (lane&15); VGPR j holds
  // K = k + j + 2*(lane>>4). Every operand fetch is a contiguous float2.
  const float* arow = Abase + (long long)a * strideA
                    + (long long)(tm * 16 + lr) * K + 2 * half;
  const float* b0 = Wbase + (long long)a * strideW
                  + (long long)(tn * 64 + lr) * K + 2 * half;
  const float* b1 = b0 + 16LL * K;
  const float* b2 = b0 + 32LL * K;
  const float* b3 = b0 + 48LL * K;

  v8f acc0 = {}, acc1 = {}, acc2 = {}, acc3 = {};

  // Prologue: load K-step 0
  v2f ac, c0, c1, c2, c3;
  ac[0] = arow[0]; ac[1] = arow[1];
  c0[0] = b0[0];   c0[1] = b0[1];
  c1[0] = b1[0];   c1[1] = b1[1];
  c2[0] = b2[0];   c2[1] = b2[1];
  c3[0] = b3[0];   c3[1] = b3[1];

  // Pipelined main loop; last step peeled (no tail overread of weights).
  for (int k = 0; k < K - 4; k += 4) {
    const int kn = k + 4;
    v2f an, n0, n1, n2, n3;
    an[0] = arow[kn]; an[1] = arow[kn + 1];
    n0[0] = b0[kn];   n0[1] = b0[kn + 1];
    n1[0] = b1[kn];   n1[1] = b1[kn + 1];
    n2[0] = b2[kn];   n2[1] = b2[kn + 1];
    n3[0] = b3[kn];   n3[1] = b3[kn + 1];

    acc0 = WMMA_F32(ac, c0, acc0);
    acc1 = WMMA_F32(ac, c1, acc1);
    acc2 = WMMA_F32(ac, c2, acc2);
    acc3 = WMMA_F32(ac, c3, acc3);

    ac = an; c0 = n0; c1 = n1; c2 = n2; c3 = n3;
  }
  acc0 = WMMA_F32(ac, c0, acc0);
  acc1 = WMMA_F32(ac, c1, acc1);
  acc2 = WMMA_F32(ac, c2, acc2);
  acc3 = WMMA_F32(ac, c3, acc3);

  // C/D layout: lane -> col (lane&15), VGPR j -> row j + 8*(lane>>4)
  float* C = Cbase + (long long)a * strideC;
  const int nb = tn * 64 + lr;
  const float bb0 = bias ? bias[nb]      : 0.0f;
  const float bb1 = bias ? bias[nb + 16] : 0.0f;
  const float bb2 = bias ? bias[nb + 32] : 0.0f;
  const float bb3 = bias ? bias[nb + 48] : 0.0f;
  #pragma unroll
  for (int j = 0; j < 8; ++j) {
    const long long m = tm * 16 + j + 8 * half;
    C[m * N + nb]      = acc0[j] + bb0;
    C[m * N + nb + 16] = acc1[j] + bb1;
    C[m * N + nb + 32] = acc2[j] + bb2;
    C[m * N + nb + 48] = acc3[j] + bb3;
  }
}

// ---------------------------------------------------------------------------
// Row softmax over the last axis of scores (A, COL, COL); one 128-thread
// block per row.
// ---------------------------------------------------------------------------
__global__ __launch_bounds__(COLN) void softmax_rows(float* __restrict__ sc)
{
  __shared__ float red[COLN];
  float* row = sc + (long long)blockIdx.x * COLN;
  const int t = threadIdx.x;

  const float x = row[t];
  red[t] = x; __syncthreads();
  for (int s = COLN / 2; s > 0; s >>= 1) {
    if (t < s) red[t] = fmaxf(red[t], red[t + s]);
    __syncthreads();
  }
  const float mx = red[0]; __syncthreads();

  const float e = expf(x - mx);
  red[t] = e; __syncthreads();
  for (int s = COLN / 2; s > 0; s >>= 1) {
    if (t < s) red[t] += red[t + s];
    __syncthreads();
  }
  row[t] = e / red[0];
}

// ---------------------------------------------------------------------------
// out[a,s,t] = sigmoid( sum_f sc[a,s,f] * W0[f,s,t] * v[a,f,t] + BIA1[s,t] )
// Not a GEMM (s on both sides of the contraction). block=(t=256),
// grid=(s=COL, a=A); scores row staged in LDS; W0/v reads coalesced on t.
// ---------------------------------------------------------------------------
__global__ __launch_bounds__(TN) void attn_out(
    const float* __restrict__ sc, const float* __restrict__ W0,
    const float* __restrict__ v,  const float* __restrict__ BIA1,
    float* __restrict__ outb)
{
  const int s = blockIdx.x, a = blockIdx.y, t = threadIdx.x;
  __shared__ float sr[COLN];
  if (t < COLN) sr[t] = sc[((long long)a * COLN + s) * COLN + t];
  __syncthreads();

  const float* w  = W0 + (long long)s * TN + t;                 // f-stride COL*TN
  const float* vv = v + (long long)a * COLN * TN + t;           // f-stride TN
  float acc = 0.0f;
  for (int f = 0; f < COLN; ++f)
    acc = fmaf(sr[f] * w[(long long)f * COLN * TN], vv[(long long)f * TN], acc);

  const float val = acc + BIA1[s * TN + t];
  outb[((long long)a * COLN + s) * TN + t] = 1.0f / (1.0f + expf(-val));
}

// ---------------------------------------------------------------------------
// LayerNorm over t (256) per (a,s) row; writes v_cur, the enc slot, and
// (last layer) the v output.
// ---------------------------------------------------------------------------
__global__ __launch_bounds__(TN) void layernorm_k(
    const float* __restrict__ x, const float* __restrict__ g,
    const float* __restrict__ b, float* __restrict__ v_cur,
    float* __restrict__ enc, int slot, float* __restrict__ out_v)
{
  const int row = blockIdx.x;            // a*COL + s
  const int a = row >> 7, s = row & (COLN - 1);
  const int t = threadIdx.x;
  __shared__ float red[TN];

  const float xv = x[(long long)row * TN + t];
  red[t] = xv; __syncthreads();
  for (int st = TN / 2; st > 0; st >>= 1) {
    if (t < st) red[t] += red[t + st];
    __syncthreads();
  }
  const float mu = red[0] * (1.0f / TN); __syncthreads();

  const float d = xv - mu;
  red[t] = d * d; __syncthreads();
  for (int st = TN / 2; st > 0; st >>= 1) {
    if (t < st) red[t] += red[t + st];
    __syncthreads();
  }
  const float var = red[0] * (1.0f / TN);
  const float y   = d * rsqrtf(var + 1e-5f) * g[t] + b[t];

  v_cur[(long long)row * TN + t] = y;
  enc[(((long long)a * (LN + 1) + slot) * COLN + s) * TN + t] = y;
  if (out_v) out_v[(long long)row * TN + t] = y;
}

// v_cur = v_final; enc[:,0] = v_final
__global__ __launch_bounds__(TN) void init_k(
    const float* __restrict__ vf, float* __restrict__ v_cur,
    float* __restrict__ enc)
{
  const int row = blockIdx.x;
  const int a = row >> 7, s = row & (COLN - 1);
  const int t = threadIdx.x;
  const float x = vf[(long long)row * TN + t];
  v_cur[(long long)row * TN + t] = x;
  enc[(((long long)a * (LN + 1)) * COLN + s) * TN + t] = x;
}

extern "C" void kernel_launch(void* const* d_in, const int* in_sizes, int n_in,
                              void* d_out, int out_size, void* d_ws, size_t ws_size,
                              hipStream_t stream)
{
  (void)in_sizes; (void)n_in; (void)out_size; (void)ws_size;

  const float* v_final = (const float*)d_in[0];
  // d_in[1] batch_x_encoder: every enc slot is rewritten -> unused
  const float* WW0   = (const float*)d_in[2];
  const float* BIA1  = (const float*)d_in[3];
  const float* pW    = (const float*)d_in[4];
  const float* pB    = (const float*)d_in[5];
  const float* qW    = (const float*)d_in[6];
  const float* qB    = (const float*)d_in[7];
  const float* projW = (const float*)d_in[8];
  const float* projB = (const float*)d_in[9];
  const float* lnG   = (const float*)d_in[10];
  const float* lnB   = (const float*)d_in[11];
  const float* qWl   = (const float*)d_in[12];
  const float* qBl   = (const float*)d_in[13];

  // Output layout (tuple flattened): v, enc, q_next, WW0, BIA1
  const long long V_SZ   = (long long)AN * COLN * TN;          // 262144
  const long long ENC_SZ = (long long)AN * (LN + 1) * COLN * TN;
  const long long W0_SZ  = (long long)COLN * COLN * TN;
  const long long B1_SZ  = (long long)COLN * TN;
  float* out_v    = (float*)d_out;
  float* out_enc  = out_v + V_SZ;
  float* out_qn   = out_enc + ENC_SZ;
  float* out_ww0  = out_qn + V_SZ;
  float* out_bia1 = out_ww0 + W0_SZ;

  // Workspace (all regions written before read every call)
  float* ws    = (float*)d_ws;
  float* v_cur = ws;                 // A*COL*T
  float* qb    = v_cur + V_SZ;       // A*COL*T
  float* pb    = qb + V_SZ;          // A*COL*T (also reused as pre-LN buf)
  float* sc    = pb + V_SZ;          // A*COL*COL
  float* outb  = sc + (long long)AN * COLN * COLN;  // A*COL*T

  const long long CS = (long long)COLN * TN;   // activation batch stride
  const long long SS = (long long)COLN * COLN; // scores batch stride

  // Passthrough outputs + init
  hipMemcpyAsync(out_ww0, WW0, W0_SZ * sizeof(float),
                 hipMemcpyDeviceToDevice, stream);
  hipMemcpyAsync(out_bia1, BIA1, B1_SZ * sizeof(float),
                 hipMemcpyDeviceToDevice, stream);
  init_k<<<AN * COLN, TN, 0, stream>>>(v_final, v_cur, out_enc);

  // GEMM grids: one wave per 16x64 strip, 4 waves (128 thr) per block
  const int blocksAct = AN * (COLN / 16) * (TN / 64) / 4;   // 64
  const int blocksSc  = AN * (COLN / 16) * (COLN / 64) / 4; // 32

  for (int i = 0; i < LN; ++i) {
    const float* qWi = qW + (long long)i * TN * TN;
    const float* pWi = pW + (long long)i * TN * TN;
    const float* jWi = projW + (long long)i * TN * TN;

    // q = v @ qW^T + qB ; p = v @ pW^T + pB
    gemm_wmma_f32<<<blocksAct, 128, 0, stream>>>(
        v_cur, CS, qWi, 0, qB + i * TN, qb, CS, TN, TN, COLN / 16, TN / 64);
    gemm_wmma_f32<<<blocksAct, 128, 0, stream>>>(
        v_cur, CS, pWi, 0, pB + i * TN, pb, CS, TN, TN, COLN / 16, TN / 64);

    // scores = p @ q^T (batched weights: strideW = CS)
    gemm_wmma_f32<<<blocksSc, 128, 0, stream>>>(
        pb, CS, qb, CS, nullptr, sc, SS, COLN, TN, COLN / 16, COLN / 64);
    softmax_rows<<<AN * COLN, COLN, 0, stream>>>(sc);

    // out = sigmoid(einsum(asf,fst,aft->ast) + BIA1)
    attn_out<<<dim3(COLN, AN), TN, 0, stream>>>(sc, WW0, v_cur, BIA1, outb);

    // v_pre = out @ projW^T + projB   (into pb, which is free now)
    gemm_wmma_f32<<<blocksAct, 128, 0, stream>>>(
        outb, CS, jWi, 0, projB + i * TN, pb, CS, TN, TN, COLN / 16, TN / 64);

    // v = layernorm(v_pre); enc[:, i+1] = v; last layer also -> out_v
    layernorm_k<<<AN * COLN, TN, 0, stream>>>(
        pb, lnG + i * TN, lnB + i * TN, v_cur, out_enc, i + 1,
        (i == LN - 1) ? out_v : nullptr);
  }

  // q_next = v @ qWl^T + qBl
  gemm_wmma_f32<<<blocksAct, 128, 0, stream>>>(
      v_cur, CS, qWl, 0, qBl, out_qn, CS, TN, TN, COLN / 16, TN / 64);
}